// FeedForwardQuantum_65481071404952
// MI455X (gfx1250) — compile-verified
//
#include <hip/hip_runtime.h>

// CDNA5 (gfx1250) fused "quantum FF" kernel:
//   out[m, 0:8] = relu( (cos(x[m,:])*cos(theta)) @ W1^T + b1 ) @ W2^T + b2
// M = 65536 rows, K1 = 8, F = 2048, all fp32 via V_WMMA_F32_16X16X4_F32.
// Each wave fuses both GEMMs for TWO 16-row tiles, sharing all weight/bias
// LDS traffic between them.

typedef __attribute__((ext_vector_type(2))) float v2f;
typedef __attribute__((ext_vector_type(8))) float v8f;

#define NQ 8

__device__ __forceinline__ v8f wmma4(v2f a, v2f b, v8f c) {
  // (neg_a, A, neg_b, B, c_mod, C, reuse_a, reuse_b)
  return __builtin_amdgcn_wmma_f32_16x16x4_f32(false, a, false, b, (short)0, c,
                                               false, false);
}

// Exactly one v_max_num_f32 (avoids the canonicalize the compiler adds to fmaxf)
__device__ __forceinline__ float relu1(float x) {
  float r;
  asm("v_max_num_f32 %0, 0, %1" : "=v"(r) : "v"(x));
  return r;
}

// Half-wave register swap: on return  a = {a.lo | b.lo->hi},  b = {a.hi->lo | b.hi}.
__device__ __forceinline__ void halfswap(float& a, float& b, int h) {
#if __has_builtin(__builtin_amdgcn_permlane16_swap)
  typedef __attribute__((ext_vector_type(2))) unsigned int v2u;
  v2u r = __builtin_amdgcn_permlane16_swap(__float_as_uint(a),
                                           __float_as_uint(b),
                                           false, false);
  a = __uint_as_float(r.x);
  b = __uint_as_float(r.y);
#else
  float sa = __shfl_xor(a, 16, 32);
  float sb = __shfl_xor(b, 16, 32);
  float na = h ? sb : a;
  float nb = h ? b : sa;
  a = na;
  b = nb;
#endif
}

extern "C" __global__ __launch_bounds__(256)
void ffq_fused(const float* __restrict__ x,
               const float* __restrict__ theta,
               const float* __restrict__ w1,   // [F, 8] row-major
               const float* __restrict__ b1,   // [F]
               const float* __restrict__ w2,   // [8, F] row-major
               const float* __restrict__ b2,   // [8]
               float* __restrict__ out,        // [M, 8]
               int M, int F)
{
  extern __shared__ float smem[];
  float* sW1 = smem;                     // F*8 floats
  float* sW2 = smem + (size_t)F * 8;     // 8*F floats
  float* sB1 = smem + (size_t)F * 16;    // F floats

  // ---- cooperative stage of weights into LDS (136 KB of the 320 KB WGP) ----
  {
    const int tid = threadIdx.x;
    const int nw  = (F * 8) >> 2;   // float4 count for each weight matrix
    const int nb  = F >> 2;
    const float4* g1 = (const float4*)w1;
    const float4* g2 = (const float4*)w2;
    const float4* g3 = (const float4*)b1;
    float4* s1 = (float4*)sW1;
    float4* s2 = (float4*)sW2;
    float4* s3 = (float4*)sB1;
    for (int i = tid; i < nw; i += blockDim.x) s1[i] = g1[i];
    for (int i = tid; i < nw; i += blockDim.x) s2[i] = g2[i];
    for (int i = tid; i < nb; i += blockDim.x) s3[i] = g3[i];
  }
  __syncthreads();

  const int lane = threadIdx.x & 31;
  const int wave = threadIdx.x >> 5;
  const int l16  = lane & 15;
  const int h    = lane >> 4;   // half-wave id: A/B hi lanes hold K offset +2
  const int h2   = h * 2;

  // cos(theta) slots this lane feeds into the B operand of GEMM1
  const float ct0 = cosf(theta[h2]);
  const float ct1 = cosf(theta[h2 + 1]);
  const float ct2 = cosf(theta[4 + h2]);
  const float ct3 = cosf(theta[5 + h2]);

  float bv[8];
  #pragma unroll
  for (int r = 0; r < 8; ++r) bv[r] = b2[r];  // uniform -> s_loads

  const int nTiles = M >> 4;
  const int nF     = F >> 4;
  const int waveId = blockIdx.x * (blockDim.x >> 5) + wave;
  const int nWaves = gridDim.x * (blockDim.x >> 5);
  const int nPairs = (nTiles + 1) >> 1;

  // W2 A-operand row: q = l16 for q<8; rows 8..15 of C2 are discarded, so
  // lanes with l16>=8 may load duplicate (garbage) rows instead of zeros.
  const float* w2base = sW2 + (size_t)(l16 & 7) * F;

  for (int p = waveId; p < nPairs; p += nWaves) {
    int tA = 2 * p;
    int tB = 2 * p + 1;
    if (tB >= nTiles) tB = tA;   // duplicate (benign) work for ragged edge

    int rowA = tA * 16 + l16;
    int rowB = tB * 16 + l16;
    int rows[2] = {rowA, rowB};

    // Q^T (8 x 16) per tile in B-matrix layout: two k-step pairs
    v2f qA[2], qB[2];
    #pragma unroll
    for (int tt = 0; tt < 2; ++tt) {
      const float* xr = x + (size_t)rows[tt] * NQ;
      float2 xa = *(const float2*)(xr + h2);
      float2 xb = *(const float2*)(xr + 4 + h2);
      qA[tt].x = cosf(xa.x) * ct0;  qA[tt].y = cosf(xa.y) * ct1;  // q=0..3
      qB[tt].x = cosf(xb.x) * ct2;  qB[tt].y = cosf(xb.y) * ct3;  // q=4..7
    }

    v8f acc0[2], acc1[2];   // out^T accumulators, b2 folded into acc0
    #pragma unroll
    for (int tt = 0; tt < 2; ++tt) {
      #pragma unroll
      for (int r = 0; r < 8; ++r) { acc0[tt][r] = bv[r]; acc1[tt][r] = 0.0f; }
    }

    #pragma unroll 2
    for (int ft = 0; ft < nF; ++ft) {
      const int fbase = ft << 4;

      // ---- shared A operands / bias for this f-tile (row-independent) ----
      const float* w1r = sW1 + (size_t)(fbase + l16) * NQ;  // GEMM1 A, M=l16
      float2 wa1 = *(const float2*)(w1r + h2);
      float2 wb1 = *(const float2*)(w1r + 4 + h2);
      v2f a1a, a1b;
      a1a.x = wa1.x; a1a.y = wa1.y;
      a1b.x = wb1.x; a1b.y = wb1.y;

      const float* bb = sB1 + fbase + 8 * h;   // C layout: f = r + 8*h
      float4 bA = *(const float4*)(bb);
      float4 bB = *(const float4*)(bb + 4);

      const float* w2r = w2base + fbase;       // GEMM2 A, q = l16&7
      float2 t0 = *(const float2*)(w2r + h2);
      float2 t1 = *(const float2*)(w2r + 4 + h2);
      float2 t2 = *(const float2*)(w2r + 8 + h2);
      float2 t3 = *(const float2*)(w2r + 12 + h2);
      v2f A0, A1, A2, A3;
      A0.x = t0.x; A0.y = t0.y;
      A1.x = t1.x; A1.y = t1.y;
      A2.x = t2.x; A2.y = t2.y;
      A3.x = t3.x; A3.y = t3.y;

      // ---- per row-tile: GEMM1 -> relu -> layout fix -> GEMM2 ----
      #pragma unroll
      for (int tt = 0; tt < 2; ++tt) {
        v8f c1;
        c1[0] = bA.x; c1[1] = bA.y; c1[2] = bA.z; c1[3] = bA.w;
        c1[4] = bB.x; c1[5] = bB.y; c1[6] = bB.z; c1[7] = bB.w;

        c1 = wmma4(a1a, qA[tt], c1);
        c1 = wmma4(a1b, qB[tt], c1);

        float c0 = relu1(c1[0]), c1v = relu1(c1[1]), c2 = relu1(c1[2]),
              c3 = relu1(c1[3]), c4 = relu1(c1[4]), c5 = relu1(c1[5]),
              c6 = relu1(c1[6]), c7 = relu1(c1[7]);

        // C-layout -> B-operand layout: 4 v_permlane16_swap_b32
        halfswap(c0, c2, h);
        halfswap(c1v, c3, h);
        halfswap(c4, c6, h);
        halfswap(c5, c7, h);
        v2f B0, B1, B2, B3;
        B0.x = c0;  B0.y = c1v;
        B1.x = c4;  B1.y = c5;
        B2.x = c2;  B2.y = c3;
        B3.x = c6;  B3.y = c7;

        acc0[tt] = wmma4(A0, B0, acc0[tt]);
        acc1[tt] = wmma4(A1, B1, acc1[tt]);
        acc0[tt] = wmma4(A2, B2, acc0[tt]);
        acc1[tt] = wmma4(A3, B3, acc1[tt]);
      }
    }

    // C2 layout: VGPR r = q, lanes 0..15 = row; only q<8 valid -> lanes 0..15
    if (lane < 16) {
      #pragma unroll
      for (int tt = 0; tt < 2; ++tt) {
        float* o = out + (size_t)rows[tt] * NQ;
        float4 o0 = make_float4(acc0[tt][0] + acc1[tt][0],
                                acc0[tt][1] + acc1[tt][1],
                                acc0[tt][2] + acc1[tt][2],
                                acc0[tt][3] + acc1[tt][3]);
        float4 o1 = make_float4(acc0[tt][4] + acc1[tt][4],
                                acc0[tt][5] + acc1[tt][5],
                                acc0[tt][6] + acc1[tt][6],
                                acc0[tt][7] + acc1[tt][7]);
        *(float4*)o = o0;
        *(float4*)(o + 4) = o1;
      }
    }
  }
}

extern "C" void kernel_launch(void* const* d_in, const int* in_sizes, int n_in,
                              void* d_out, int out_size, void* d_ws, size_t ws_size,
                              hipStream_t stream) {
  const float* x     = (const float*)d_in[0];
  const float* theta = (const float*)d_in[1];
  const float* w1    = (const float*)d_in[2];
  const float* b1    = (const float*)d_in[3];
  const float* w2    = (const float*)d_in[4];
  const float* b2    = (const float*)d_in[5];
  float* out = (float*)d_out;

  const int M = in_sizes[0] / NQ;   // 65536 rows
  const int F = in_sizes[3];        // 2048

  const int nTiles = M / 16;                 // 4096 row tiles, 16 rows each
  const int nPairs = (nTiles + 1) / 2;       // 2 row tiles per wave
  const int wavesPerBlock = 8;               // 256 threads = 8 wave32
  int blocks = (nPairs + wavesPerBlock - 1) / wavesPerBlock;
  if (blocks > 256) blocks = 256;            // grid-stride over tile pairs

  const size_t lds = (size_t)F * 17 * sizeof(float);  // W1 + W2 + b1 = 136 KB

  ffq_fused<<<blocks, 256, lds, stream>>>(x, theta, w1, b1, w2, b2, out, M, F);
}